// SinkhornDistance_83846351552663
// MI455X (gfx1250) — compile-verified
//
#include <hip/hip_runtime.h>

// ---------------------------------------------------------------------------
// Sinkhorn distance, N=8192, D=64, EPS=0.01, 100 iters.
//
// MI455X plan:
//  - Build C = x2 + y2 - 2*X*Y^T once in exact fp32 via V_WMMA_F32_16X16X4_F32
//    (C/eps amplifies any rounding of C by 100x -> low precision unusable).
//  - 100 iterations x 2 LSE passes over the 256MB C matrix: 51.2 GB streamed
//    -> ~2.2 ms floor at 23.3 TB/s. All passes use b128 loads (512B/wave/instr)
//    and log2-domain online logsumexp (v_exp_f32 / v_log_f32 are natively
//    base-2) to stay on the bandwidth roof rather than the VALU/issue roof.
//  - Deterministic fixed-tree reductions (graph-replay safe, no float atomics).
// ---------------------------------------------------------------------------

typedef __attribute__((ext_vector_type(2))) float v2f;
typedef __attribute__((ext_vector_type(8))) float v8f;

constexpr int   N        = 8192;
constexpr int   D        = 64;
constexpr float EPS      = 0.01f;
constexpr int   MAX_ITER = 100;
constexpr float LOG_MU   = -9.01091334f;      // -ln(8192)
constexpr float K2       = 144.269504089f;    // (1/EPS) * log2(e)
constexpr float LN2      = 0.693147180560f;

constexpr int CI_ROWS  = 128;                 // rows per column-pass chunk
constexpr int NCHUNK_I = N / CI_ROWS;         // 64
constexpr int CJ_COLS  = 2048;                // cols per row-pass chunk
constexpr int NCHUNK_J = N / CJ_COLS;         // 4
constexpr int SUM_BLOCKS = N / 2;             // 4096 (2 rows per block)

// Single-instruction base-2 transcendentals (v_exp_f32 / v_log_f32)
__device__ __forceinline__ float fexp2(float x) { return __builtin_amdgcn_exp2f(x); }
__device__ __forceinline__ float flog2(float x) { return __builtin_amdgcn_logf(x); }

// Online logsumexp in log2 domain: state (m, s) represents log2(sum) = m + log2(s)
__device__ __forceinline__ void lse_push(float& m, float& s, float t) {
  float nm = fmaxf(m, t);
  s = s * fexp2(m - nm) + fexp2(t - nm);
  m = nm;
}
__device__ __forceinline__ void lse_merge(float& m, float& s, float om, float os) {
  float nm = fmaxf(m, om);
  s = s * fexp2(m - nm) + os * fexp2(om - nm);
  m = nm;
}

// --- row norms + u/v zero-init (ws is not re-poisoned between replays) ------
__global__ void norms_init(const float* __restrict__ x, const float* __restrict__ y,
                           float* __restrict__ x2, float* __restrict__ y2,
                           float* __restrict__ u, float* __restrict__ v) {
  int i = blockIdx.x * blockDim.x + threadIdx.x;        // 8192 threads
  const float4* xr = (const float4*)(x + (size_t)i * D);
  const float4* yr = (const float4*)(y + (size_t)i * D);
  float sx = 0.f, sy = 0.f;
#pragma unroll
  for (int t = 0; t < D / 4; ++t) {
    float4 a = xr[t]; sx += a.x * a.x + a.y * a.y + a.z * a.z + a.w * a.w;
    float4 b = yr[t]; sy += b.x * b.x + b.y * b.y + b.z * b.z + b.w * b.w;
  }
  x2[i] = sx; y2[i] = sy; u[i] = 0.f; v[i] = 0.f;
}

// --- C[i,j] = x2[i] + y2[j] - 2 * dot(x_i, y_j) via fp32 WMMA ----------------
// Block = 256 threads = 8 waves as 2x4 tiles -> 32x64 macro-tile of C.
// A (16x4 f32): lanes 0-15 M=lane, VGPR{0,1}=K{0,1}; lanes 16-31 M=lane-16,
// K{2,3}. B (4x16) mirrors with N<->M, so both operands load identically from
// row-major X and Y (B = Y^T).
__global__ void cost_wmma(const float* __restrict__ x, const float* __restrict__ y,
                          const float* __restrict__ x2, const float* __restrict__ y2,
                          float* __restrict__ C) {
  int lane = threadIdx.x & 31;
  int wave = threadIdx.x >> 5;
  int i0 = blockIdx.y * 32 + (wave >> 2) * 16;
  int j0 = blockIdx.x * 64 + (wave & 3) * 16;

  int m16 = lane & 15;
  int kb  = (lane >> 4) << 1;                 // 0 or 2
  const float* xr = x + (size_t)(i0 + m16) * D + kb;
  const float* yr = y + (size_t)(j0 + m16) * D + kb;

  v8f acc = {};
#pragma unroll
  for (int t = 0; t < D / 4; ++t) {           // 16 chained WMMAs, K = 64
    v2f a = *(const v2f*)(xr + t * 4);
    v2f b = *(const v2f*)(yr + t * 4);
    acc = __builtin_amdgcn_wmma_f32_16x16x4_f32(
        false, a, false, b, (short)0, acc, false, false);
  }

  int hi = lane >> 4;
  int n  = lane & 15;
  float yn = y2[j0 + n];
#pragma unroll
  for (int r = 0; r < 8; ++r) {
    int m = r + 8 * hi;
    float cval = x2[i0 + m] + yn - 2.0f * acc[r];
    C[(size_t)(i0 + m) * N + (j0 + n)] = cval;
  }
}

// --- v update, stage 1: per (row-chunk, 4-column) partial LSE ----------------
// Thread owns 4 consecutive columns (b128 loads); block covers 1024 columns.
__global__ void col_pass(const float* __restrict__ C, const float* __restrict__ u,
                         float2* __restrict__ part) {
  int j0    = blockIdx.x * 1024 + threadIdx.x * 4;   // gridDim.x = 8
  int ibase = blockIdx.y * CI_ROWS;                  // gridDim.y = 64
  const float4* col = (const float4*)(C + (size_t)ibase * N + j0);
  float m0 = -1e30f, s0 = 0.f, m1 = -1e30f, s1 = 0.f;
  float m2 = -1e30f, s2 = 0.f, m3 = -1e30f, s3 = 0.f;
  for (int r = 0; r < CI_ROWS; ++r) {
    float4 c = col[(size_t)r * (N / 4)];             // coalesced b128 stream
    float uk = u[ibase + r] * K2;                    // uniform -> scalar path
    lse_push(m0, s0, fmaf(c.x, -K2, uk));
    lse_push(m1, s1, fmaf(c.y, -K2, uk));
    lse_push(m2, s2, fmaf(c.z, -K2, uk));
    lse_push(m3, s3, fmaf(c.w, -K2, uk));
  }
  float4* p4 = (float4*)(part + (size_t)blockIdx.y * N + j0);
  p4[0] = make_float4(m0, s0, m1, s1);
  p4[1] = make_float4(m2, s2, m3, s3);
}

// --- v update, stage 2: merge 64 chunks, apply update ------------------------
__global__ void col_combine(const float2* __restrict__ part, float* __restrict__ v) {
  int j = blockIdx.x * 256 + threadIdx.x;            // 32 blocks
  float m = -1e30f, s = 0.f;
#pragma unroll 8
  for (int c = 0; c < NCHUNK_I; ++c) {
    float2 p = part[(size_t)c * N + j];
    lse_merge(m, s, p.x, p.y);
  }
  float lse = (m + flog2(s)) * LN2;                  // back to natural log
  v[j] += EPS * (LOG_MU - lse);
}

// --- u update, stage 1: wave-per-row partial LSE over a 2048-column chunk ----
// 8 waves per block -> 8 rows; each lane streams float4s (b128, 512B/wave).
__global__ void row_pass(const float* __restrict__ C, const float* __restrict__ v,
                         float2* __restrict__ part) {
  int lane  = threadIdx.x & 31;
  int wave  = threadIdx.x >> 5;
  int i     = blockIdx.x * 8 + wave;
  int jbase = blockIdx.y * CJ_COLS;
  const float4* row4 = (const float4*)(C + (size_t)i * N + jbase);
  const float4* vv4  = (const float4*)(v + jbase);
  float m0 = -1e30f, s0 = 0.f, m1 = -1e30f, s1 = 0.f;
  float m2 = -1e30f, s2 = 0.f, m3 = -1e30f, s3 = 0.f;
#pragma unroll 4
  for (int t = 0; t < CJ_COLS / 128; ++t) {          // 16 steps
    int idx = t * 32 + lane;
    float4 c  = row4[idx];
    float4 vj = vv4[idx];
    lse_push(m0, s0, (vj.x - c.x) * K2);
    lse_push(m1, s1, (vj.y - c.y) * K2);
    lse_push(m2, s2, (vj.z - c.z) * K2);
    lse_push(m3, s3, (vj.w - c.w) * K2);
  }
  lse_merge(m0, s0, m1, s1);
  lse_merge(m2, s2, m3, s3);
  lse_merge(m0, s0, m2, s2);
  // wave32 LSE reduction tree
#pragma unroll
  for (int off = 16; off > 0; off >>= 1) {
    float om = __shfl_xor(m0, off, 32);
    float os = __shfl_xor(s0, off, 32);
    lse_merge(m0, s0, om, os);
  }
  if (lane == 0) part[(size_t)blockIdx.y * N + i] = make_float2(m0, s0);
}

// --- u update, stage 2 -------------------------------------------------------
__global__ void row_combine(const float2* __restrict__ part, float* __restrict__ u) {
  int i = blockIdx.x * 256 + threadIdx.x;            // 32 blocks
  float m = -1e30f, s = 0.f;
#pragma unroll
  for (int c = 0; c < NCHUNK_J; ++c) {
    float2 p = part[(size_t)c * N + i];
    lse_merge(m, s, p.x, p.y);
  }
  float lse = (m + flog2(s)) * LN2;
  u[i] += EPS * (LOG_MU - lse);
}

// --- final distance: sum(exp((u_i+v_j-C)/eps) * C), deterministic tree -------
__global__ void plan_sum(const float* __restrict__ C, const float* __restrict__ u,
                         const float* __restrict__ v, float* __restrict__ part1) {
  int i0 = blockIdx.x * 2;                           // 2 full rows per block
  const float4* vv4 = (const float4*)v;
  float acc = 0.f;
#pragma unroll
  for (int r = 0; r < 2; ++r) {
    int i = i0 + r;
    float uk = u[i] * K2;                            // uniform per row
    const float4* row4 = (const float4*)(C + (size_t)i * N);
#pragma unroll 2
    for (int t = 0; t < N / 1024; ++t) {             // 8 steps of b128
      int idx = t * 256 + threadIdx.x;
      float4 c  = row4[idx];
      float4 vj = vv4[idx];
      acc += fexp2(fmaf(vj.x - c.x, K2, uk)) * c.x;
      acc += fexp2(fmaf(vj.y - c.y, K2, uk)) * c.y;
      acc += fexp2(fmaf(vj.z - c.z, K2, uk)) * c.z;
      acc += fexp2(fmaf(vj.w - c.w, K2, uk)) * c.w;
    }
  }
  __shared__ float red[256];
  red[threadIdx.x] = acc;
  __syncthreads();
  for (int off = 128; off > 0; off >>= 1) {
    if (threadIdx.x < off) red[threadIdx.x] += red[threadIdx.x + off];
    __syncthreads();
  }
  if (threadIdx.x == 0) part1[blockIdx.x] = red[0];
}

__global__ void final_sum(const float* __restrict__ part1, float* __restrict__ out) {
  __shared__ double red[256];
  double acc = 0.0;
  for (int t = threadIdx.x; t < SUM_BLOCKS; t += 256) acc += (double)part1[t];
  red[threadIdx.x] = acc;
  __syncthreads();
  for (int off = 128; off > 0; off >>= 1) {
    if (threadIdx.x < off) red[threadIdx.x] += red[threadIdx.x + off];
    __syncthreads();
  }
  if (threadIdx.x == 0) out[0] = (float)(red[0] / (double)N);
}

extern "C" void kernel_launch(void* const* d_in, const int* in_sizes, int n_in,
                              void* d_out, int out_size, void* d_ws, size_t ws_size,
                              hipStream_t stream) {
  const float* x = (const float*)d_in[0];
  const float* y = (const float*)d_in[1];

  // Workspace layout (floats): C | u | v | x2 | y2 | partials (4 MB)
  float*  ws      = (float*)d_ws;
  float*  C       = ws;
  float*  u       = ws + (size_t)N * N;
  float*  v       = u + N;
  float*  x2      = v + N;
  float*  y2      = x2 + N;
  float2* part    = (float2*)(y2 + N);               // NCHUNK_I * N float2
  float*  sumpart = (float*)part;                    // reused after iterations

  norms_init<<<N / 256, 256, 0, stream>>>(x, y, x2, y2, u, v);
  cost_wmma<<<dim3(N / 64, N / 32), 256, 0, stream>>>(x, y, x2, y2, C);

  for (int it = 0; it < MAX_ITER; ++it) {
    col_pass   <<<dim3(N / 1024, NCHUNK_I), 256, 0, stream>>>(C, u, part);
    col_combine<<<N / 256, 256, 0, stream>>>(part, v);
    row_pass   <<<dim3(N / 8, NCHUNK_J), 256, 0, stream>>>(C, v, part);
    row_combine<<<N / 256, 256, 0, stream>>>(part, u);
  }

  plan_sum <<<SUM_BLOCKS, 256, 0, stream>>>(C, u, v, sumpart);
  final_sum<<<1, 256, 0, stream>>>(sumpart, (float*)d_out);
}